// InfoNCELoss_58669253263549
// MI455X (gfx1250) — compile-verified
//
#include <hip/hip_runtime.h>
#include <math.h>

#define N_PRED 12
#define B_SZ   256
#define N_NEG  128
#define D_DIM  512

#define LOG2E 1.4426950408889634f

typedef float v2f __attribute__((ext_vector_type(2)));
typedef float v4f __attribute__((ext_vector_type(4)));
typedef float v8f __attribute__((ext_vector_type(8)));

__global__ void infonce_zero_kernel(float* out) {
    if (threadIdx.x < 1 + N_PRED) out[threadIdx.x] = 0.0f;
}

__launch_bounds__(256)
__global__ void InfoNCELoss_58669253263549_kernel(const float* __restrict__ lp,
                                                  const float* __restrict__ ps,
                                                  const int*   __restrict__ time_idx,
                                                  const int*   __restrict__ batch_idx,
                                                  float*       __restrict__ out) {
    const int kb   = blockIdx.x;      // 0 .. N_PRED*B_SZ-1
    const int k    = kb / B_SZ;       // prediction step
    const int b    = kb % B_SZ;       // batch element
    const int tid  = threadIdx.x;
    const int wave = tid >> 5;        // 0..7
    const int lane = tid & 31;        // lane within wave32

    __shared__ __align__(16) float p_lds[D_DIM];  // pred * log2(e) for this (k,b)
    __shared__ float red[256];                    // fk_pos reduction
    __shared__ float wsum[16];                    // per half-wave neg sums
    __shared__ float wmax[16];                    // per half-wave neg maxes
    __shared__ float s_fkpos;

    const float* pred = lp + ((size_t)(k + 1) * B_SZ + b) * D_DIM;
    const float* pos  = ps + ((size_t)k       * B_SZ + b) * D_DIM;

    // ---- Phase 0: stage pred*log2(e) into LDS; partial fk_pos = sum_d exp(p_d*pos_d)
    //      exp(p*x) == exp2((p*log2e)*x), so the scale is paid once per pred element.
    float pp = 0.0f;
#pragma unroll
    for (int j = 0; j < D_DIM / 256; ++j) {
        int d = tid + j * 256;
        float pvs = pred[d] * LOG2E;
        p_lds[d] = pvs;
        pp += __builtin_amdgcn_exp2f(pvs * pos[d]);
    }
    red[tid] = pp;
    __syncthreads();

    // block tree-reduce fk_pos
    for (int s = 128; s > 0; s >>= 1) {
        if (tid < s) red[tid] += red[tid + s];
        __syncthreads();
    }
    if (tid == 0) s_fkpos = red[0];

    // ---- Phase 1: negatives. Wave w owns negatives [16w, 16w+16).
    // A-matrix row r is fed by lane r (as K0,K1) and lane r+16 (as K2,K3);
    // with B == ones the WMMA computes row sums on the matrix pipe, so lane r
    // covers d in [0,256) and lane r+16 covers d in [256,512) of the same row.
    const int r  = lane & 15;
    const int hi = lane >> 4;                  // 0: low half of D, 1: high half
    const int n  = wave * 16 + r;              // negative index 0..127
    const size_t iofs = ((size_t)k * B_SZ + b) * N_NEG + n;
    const int t = time_idx[iofs];
    const int c = batch_idx[iofs];
    // bank[0] = lp[0], bank[j>=1] = ps[j-1]
    const float* rowp = (t == 0) ? (lp + (size_t)c * D_DIM)
                                 : (ps + ((size_t)(t - 1) * B_SZ + c) * D_DIM);
    const float* ep     = rowp  + hi * (D_DIM / 2);
    const float* plocal = p_lds + hi * (D_DIM / 2);

    v8f acc0 = {0.f, 0.f, 0.f, 0.f, 0.f, 0.f, 0.f, 0.f};
    v8f acc1 = {0.f, 0.f, 0.f, 0.f, 0.f, 0.f, 0.f, 0.f};
    const v2f bones = {1.0f, 1.0f};

#pragma unroll 4
    for (int i = 0; i < (D_DIM / 2) / 4; ++i) {  // 64 iterations
        v4f x  = *(const v4f*)(ep     + i * 4);  // global_load_b128 (gathered row)
        v4f pv = *(const v4f*)(plocal + i * 4);  // ds_load_b128 (broadcast pred)
        v2f a0, a1;
        a0.x = __builtin_amdgcn_exp2f(x.x * pv.x);
        a0.y = __builtin_amdgcn_exp2f(x.y * pv.y);
        a1.x = __builtin_amdgcn_exp2f(x.z * pv.z);
        a1.y = __builtin_amdgcn_exp2f(x.w * pv.w);
        // row-sum via matrix pipe: D = A * ones + C
        acc0 = __builtin_amdgcn_wmma_f32_16x16x4_f32(false, a0, false, bones,
                                                     (short)0, acc0, false, false);
        acc1 = __builtin_amdgcn_wmma_f32_16x16x4_f32(false, a1, false, bones,
                                                     (short)0, acc1, false, false);
    }

    // C/D layout: lane 0 holds rows 0..7 (one per VGPR), lane 16 rows 8..15.
    if ((lane & 15) == 0) {
        float s = 0.0f, m = -INFINITY;
#pragma unroll
        for (int j = 0; j < 8; ++j) {
            float v = acc0[j] + acc1[j];
            s += v;
            m = fmaxf(m, v);
        }
        wsum[wave * 2 + hi] = s;
        wmax[wave * 2 + hi] = m;
    }
    __syncthreads();

    // ---- Phase 2: finalize per (k,b), accumulate globally.
    if (tid == 0) {
        float S = 0.0f, M = -INFINITY;
#pragma unroll
        for (int i = 0; i < 16; ++i) {
            S += wsum[i];
            M = fmaxf(M, wmax[i]);
        }
        const float fp = s_fkpos;
        const float per_step = __logf(fp / (fp + S));
        atomicAdd(&out[0], per_step * (1.0f / ((float)(N_NEG + 1) * N_PRED * B_SZ)));
        if (fp > M) atomicAdd(&out[1 + k], 1.0f);
    }
}

extern "C" void kernel_launch(void* const* d_in, const int* in_sizes, int n_in,
                              void* d_out, int out_size, void* d_ws, size_t ws_size,
                              hipStream_t stream) {
    const float* lp = (const float*)d_in[0];   // latent_predictions (13,256,512) f32
    const float* ps = (const float*)d_in[1];   // positive_samples   (12,256,512) f32
    const int*   ti = (const int*)d_in[2];     // time_idx  (12,256,128) i32
    const int*   bi = (const int*)d_in[3];     // batch_idx (12,256,128) i32
    float* out = (float*)d_out;                // [loss, correct_predictions(12)]

    infonce_zero_kernel<<<dim3(1), dim3(32), 0, stream>>>(out);
    InfoNCELoss_58669253263549_kernel<<<dim3(N_PRED * B_SZ), dim3(256), 0, stream>>>(
        lp, ps, ti, bi, out);
}